// RoutingAttention_41532333752833
// MI455X (gfx1250) — compile-verified
//
#include <hip/hip_runtime.h>

// Problem constants (match setup_inputs: B=2, L=4096, D=256, H=8, K=sqrt(L)=64)
#define B_ 2
#define L_ 4096
#define D_ 256
#define H_ 8
#define K_ 64
#define N_ (B_ * L_)   // 8192 points
#define ITERS_ 10

typedef __attribute__((ext_vector_type(2))) float v2f;
typedef __attribute__((ext_vector_type(4))) float v4f;
typedef __attribute__((ext_vector_type(8))) float v8f;

// ---------------------------------------------------------------------------
// 1) Projection: R[b] = qk[b] (L x D) @ W_R[b] (D x D), fp32 WMMA 16x16x4.
//    One wave computes one 16x16 output tile; K-loop of 64 v_wmma_f32_16x16x4_f32.
//    A layout (16x4 f32): lane = row M (0-15 twice), VGPR0/1 = K pair,
//    lanes 0-15 hold K=0,1 and lanes 16-31 hold K=2,3.
//    B layout (4x16 f32): mirrored with lane = col N.
//    C/D (16x16 f32): VGPR r = row m0+r (lanes 0-15) / m0+r+8 (lanes 16-31).
// ---------------------------------------------------------------------------
__global__ __launch_bounds__(256) void proj_wmma(const float* __restrict__ qk,
                                                 const float* __restrict__ W,
                                                 float* __restrict__ R) {
  const int lane = threadIdx.x & 31;
  const int wave = (blockIdx.x * 256 + threadIdx.x) >> 5;
  const int tilesN = D_ / 16;                // 16
  const int tilesM = L_ / 16;                // 256
  const int perBatch = tilesM * tilesN;      // 4096
  const int b  = wave / perBatch;
  const int t  = wave % perBatch;
  const int m0 = (t / tilesN) * 16;
  const int n0 = (t % tilesN) * 16;

  const float* A  = qk + (size_t)b * L_ * D_;
  const float* Bw = W  + (size_t)b * D_ * D_;
  float*       Ro = R  + (size_t)b * L_ * D_;

  const int half = lane >> 4;   // 0: lanes 0-15 (K=0,1) | 1: lanes 16-31 (K=2,3)
  const int l16  = lane & 15;

  v8f acc = {};
  for (int k0 = 0; k0 < D_; k0 += 4) {
    const int ka = k0 + half * 2;
    v2f av, bv;
    av.x = A[(size_t)(m0 + l16) * D_ + ka];
    av.y = A[(size_t)(m0 + l16) * D_ + ka + 1];
    bv.x = Bw[(size_t)ka * D_ + (n0 + l16)];
    bv.y = Bw[(size_t)(ka + 1) * D_ + (n0 + l16)];
    acc = __builtin_amdgcn_wmma_f32_16x16x4_f32(false, av, false, bv,
                                                (short)0, acc, false, false);
  }
#pragma unroll
  for (int r = 0; r < 8; ++r) {
    const int m = m0 + r + half * 8;
    Ro[(size_t)m * D_ + (n0 + l16)] = acc[r];
  }
}

// ---------------------------------------------------------------------------
// 2) Deterministic centroid init: strided sample of projected points.
// ---------------------------------------------------------------------------
__global__ void init_cent(const float* __restrict__ R, float* __restrict__ C) {
  const int idx = blockIdx.x * blockDim.x + threadIdx.x;   // 16384 threads
  const int k = idx >> 8;                                  // cluster
  const int d = idx & (D_ - 1);                            // dim
  const int p = k * (N_ / K_) + 17;                        // < 8192
  C[idx] = R[(size_t)p * D_ + d];
}

__global__ void zero_acc(float* __restrict__ Csum, float* __restrict__ Ccnt) {
  const int idx = blockIdx.x * blockDim.x + threadIdx.x;
  if (idx < K_ * D_) Csum[idx] = 0.f;
  if (idx < K_)      Ccnt[idx] = 0.f;
}

// ---------------------------------------------------------------------------
// 3) Lloyd iteration body: assign each point to nearest centroid (centroids
//    cached in 64 KB LDS; 64 fp32 accumulators per thread, fully unrolled),
//    then (optionally) aggregate per-block sums in the SAME LDS buffer
//    (barrier-separated reuse) before one global atomic pass.
// ---------------------------------------------------------------------------
__global__ __launch_bounds__(256) void assign_acc(const float* __restrict__ R,
                                                  const float* __restrict__ C,
                                                  int* __restrict__ cid,
                                                  float* __restrict__ Csum,
                                                  float* __restrict__ Ccnt,
                                                  int doAccum) {
  __shared__ float Cl[K_ * D_];   // 64 KB: centroids, then reused as sum buffer
  const int t = threadIdx.x;
  for (int i = t; i < K_ * D_; i += 256) Cl[i] = C[i];
  __syncthreads();

  const int p = blockIdx.x * 256 + t;         // one point per thread, exact fit
  const float* x = R + (size_t)p * D_;

  float acc[K_];
#pragma unroll
  for (int k = 0; k < K_; ++k) acc[k] = 0.f;

  for (int d = 0; d < D_; ++d) {
    const float xv = x[d];                    // LDS reads broadcast per wave
#pragma unroll
    for (int k = 0; k < K_; ++k) {
      const float diff = xv - Cl[k * D_ + d];
      acc[k] = fmaf(diff, diff, acc[k]);
    }
  }

  float best = acc[0];
  int bk = 0;
#pragma unroll
  for (int k = 1; k < K_; ++k)
    if (acc[k] < best) { best = acc[k]; bk = k; }
  cid[p] = bk;

  if (doAccum) {
    __syncthreads();                          // all lanes done reading Cl
    for (int i = t; i < K_ * D_; i += 256) Cl[i] = 0.f;
    __syncthreads();
    for (int d = 0; d < D_; ++d) atomicAdd(&Cl[bk * D_ + d], x[d]);
    atomicAdd(&Ccnt[bk], 1.0f);
    __syncthreads();
    for (int i = t; i < K_ * D_; i += 256) atomicAdd(&Csum[i], Cl[i]);
  }
}

__global__ void update_cent(float* __restrict__ C,
                            const float* __restrict__ Csum,
                            const float* __restrict__ Ccnt) {
  const int idx = blockIdx.x * blockDim.x + threadIdx.x;   // 16384
  const float cnt = Ccnt[idx >> 8];
  if (cnt > 0.f) C[idx] = Csum[idx] / cnt;                 // else keep old
}

// ---------------------------------------------------------------------------
// 4) Mask emission: out[b,h,i,j] = (cid[b,i]==cid[b,j]) ? 0 : -10000.
//    One block per (b,i); cid row in LDS; 16 values/thread as four b128
//    NON-TEMPORAL stores, replicated over 8 heads. 1 GiB streaming write
//    (5x the 192 MB L2) -> NT hints keep it from evicting the k-means set.
// ---------------------------------------------------------------------------
__global__ __launch_bounds__(256) void mask_out(const int* __restrict__ cid,
                                                float* __restrict__ out) {
  const int bi = blockIdx.x;            // b*L + i
  const int b  = bi >> 12;
  const int i  = bi & (L_ - 1);
  __shared__ int crow[L_];              // 16 KB
  const int* cb = cid + (size_t)b * L_;
  for (int j = threadIdx.x; j < L_; j += 256) crow[j] = cb[j];
  __syncthreads();

  const int ci = crow[i];
  const int j0 = threadIdx.x * 16;      // 256*16 = 4096 exactly
  v4f v[4];
#pragma unroll
  for (int q = 0; q < 4; ++q) {
    v[q].x = (crow[j0 + q * 4 + 0] == ci) ? 0.f : -10000.f;
    v[q].y = (crow[j0 + q * 4 + 1] == ci) ? 0.f : -10000.f;
    v[q].z = (crow[j0 + q * 4 + 2] == ci) ? 0.f : -10000.f;
    v[q].w = (crow[j0 + q * 4 + 3] == ci) ? 0.f : -10000.f;
  }
  const size_t rowBase = ((size_t)b * H_) * (size_t)L_ * L_ + (size_t)i * L_;
#pragma unroll
  for (int h = 0; h < H_; ++h) {
    v4f* dst = (v4f*)(out + rowBase + (size_t)h * L_ * L_ + j0);
#pragma unroll
    for (int q = 0; q < 4; ++q) __builtin_nontemporal_store(v[q], dst + q);
  }
}

// ---------------------------------------------------------------------------
extern "C" void kernel_launch(void* const* d_in, const int* in_sizes, int n_in,
                              void* d_out, int out_size, void* d_ws, size_t ws_size,
                              hipStream_t stream) {
  (void)in_sizes; (void)n_in; (void)out_size; (void)ws_size;
  const float* qk = (const float*)d_in[0];   // (B, L, D) fp32
  const float* W  = (const float*)d_in[1];   // (B, D, D) fp32
  float* out = (float*)d_out;                // (B, H, L, L) fp32

  char* ws = (char*)d_ws;
  float* R    = (float*)ws;  ws += (size_t)N_ * D_ * sizeof(float);   // 8 MB
  float* C    = (float*)ws;  ws += (size_t)K_ * D_ * sizeof(float);   // 64 KB
  float* Csum = (float*)ws;  ws += (size_t)K_ * D_ * sizeof(float);   // 64 KB
  float* Ccnt = (float*)ws;  ws += (size_t)K_ * sizeof(float);
  int*   cid  = (int*)ws;                                             // 32 KB

  // Projection: 8192 waves (one 16x16 tile each) = 1024 blocks of 8 waves.
  proj_wmma<<<(B_ * (L_ / 16) * (D_ / 16)) / 8, 256, 0, stream>>>(qk, W, R);

  init_cent<<<(K_ * D_) / 256, 256, 0, stream>>>(R, C);

  for (int it = 0; it < ITERS_; ++it) {
    zero_acc<<<(K_ * D_ + 255) / 256, 256, 0, stream>>>(Csum, Ccnt);
    assign_acc<<<N_ / 256, 256, 0, stream>>>(R, C, cid, Csum, Ccnt, 1);
    update_cent<<<(K_ * D_) / 256, 256, 0, stream>>>(C, Csum, Ccnt);
  }
  // Final assignment with converged centroids (matches reference's last _assign).
  assign_acc<<<N_ / 256, 256, 0, stream>>>(R, C, cid, Csum, Ccnt, 0);

  // 1 GiB mask write: 8192 blocks, one (b, i) row each, NT b128 stores.
  mask_out<<<N_, 256, 0, stream>>>(cid, out);
}